// GraphConv_4028679324256
// MI455X (gfx1250) — compile-verified
//
#include <hip/hip_runtime.h>

#define N_NODES 100000
#define N_EDGES 3200000
#define D_IN    256
#define D_OUT   128

typedef __attribute__((ext_vector_type(2))) float v2f;
typedef __attribute__((ext_vector_type(8))) float v8f;

// ---------------------------------------------------------------------------
// Kernel 1: zero the degree counters (in ws) and the output accumulator.
// ---------------------------------------------------------------------------
__global__ void gc_zero_kernel(unsigned* __restrict__ outdeg,
                               unsigned* __restrict__ indeg,
                               float* __restrict__ out) {
    int i = blockIdx.x * blockDim.x + threadIdx.x;
    if (i < N_NODES * D_OUT) out[i] = 0.0f;
    if (i < N_NODES) { outdeg[i] = 0u; indeg[i] = 0u; }
}

// ---------------------------------------------------------------------------
// Kernel 2: integer degree counts (deterministic, u32 atomics).
// ---------------------------------------------------------------------------
__global__ void gc_degree_kernel(const int* __restrict__ src,
                                 const int* __restrict__ dst,
                                 unsigned* __restrict__ outdeg,
                                 unsigned* __restrict__ indeg) {
    int e = blockIdx.x * blockDim.x + threadIdx.x;
    if (e < N_EDGES) {
        atomicAdd(&outdeg[src[e]], 1u);
        atomicAdd(&indeg[dst[e]], 1u);
    }
}

// ---------------------------------------------------------------------------
// Kernel 3: h = diag(outdeg^-1/2) * (feat @ W)  via V_WMMA_F32_16X16X4_F32.
//
// One block = 16 rows x 128 cols. 8 waves; wave w owns a 16-wide N-tile.
// The 16x256 f32 A tile is DMA'd into LDS with GLOBAL_LOAD_ASYNC_TO_LDS_B128
// (ASYNCcnt-tracked, no VGPR staging), then consumed as WMMA A-fragments via
// ds_load_b64. The out-degree row scale is applied to the C tile at store
// time (algebraically identical to pre-scaling A, lets the DMA be a raw copy).
//
// LDS row padding = +4 floats (ROW=260): keeps every async b128 store 16B
// aligned and makes the A-fragment reads bank-conflict-free (lane bank stride
// 4, 16 distinct rows).
//
// EXEC is all-ones throughout (WMMA requirement): 6250*16 == 100000 exactly.
// ---------------------------------------------------------------------------
__global__ __launch_bounds__(256) void gc_gemm_kernel(
        const float* __restrict__ feat,
        const float* __restrict__ W,
        const unsigned* __restrict__ outdeg,
        float* __restrict__ h) {
    constexpr int ROW = D_IN + 4;          // 260 floats: 16B-aligned, bank-safe
    __shared__ float As[16 * ROW];

    const int tileM = blockIdx.x;          // 0..6249
    const int t = threadIdx.x;

    // Async-stage the 16x256 A tile: 1024 x b128 chunks, 4 per thread.
    const unsigned long long fbase = (unsigned long long)feat;
    #pragma unroll
    for (int j = 0; j < 4; ++j) {
        int idx = t + 256 * j;             // b128 chunk id, 0..1023
        int m   = idx >> 6;                // tile row 0..15
        int k   = (idx & 63) << 2;         // col 0..252 step 4
        unsigned goff = (unsigned)(((tileM * 16 + m) * D_IN + k) * 4);
        unsigned ldsa = (unsigned)(uintptr_t)&As[m * ROW + k];
        asm volatile("global_load_async_to_lds_b128 %0, %1, %2"
                     :: "v"(ldsa), "v"(goff), "s"(fbase) : "memory");
    }
    asm volatile("s_wait_asynccnt 0" ::: "memory");
    __syncthreads();

    const int lane = t & 31;
    const int wv   = t >> 5;               // N-tile index 0..7
    const int m    = lane & 15;            // A row within tile
    const int kb   = (lane >> 4) << 1;     // lanes 16-31 hold K+2,K+3
    const int n    = wv * 16 + (lane & 15);

    v8f acc = {};
    #pragma unroll 8
    for (int kk = 0; kk < D_IN; kk += 4) {
        // A fragment: lane holds A[m][kk+kb], A[m][kk+kb+1]  (ds_load_b64)
        v2f a = *(const v2f*)&As[m * ROW + kk + kb];
        // B fragment: lane holds W[kk+kb][n], W[kk+kb+1][n]
        v2f b;
        b.x = W[(kk + kb)     * D_OUT + n];
        b.y = W[(kk + kb + 1) * D_OUT + n];
        acc = __builtin_amdgcn_wmma_f32_16x16x4_f32(
                  false, a, false, b, (short)0, acc, false, false);
    }

    // C/D layout: VGPR r -> M=r (lanes 0-15), M=8+r (lanes 16-31); N = lane&15.
    // Apply the out-degree row scale here.
    const int rbase = tileM * 16 + ((lane >> 4) << 3);
    #pragma unroll
    for (int r = 0; r < 8; ++r) {
        float sc = 1.0f / sqrtf(fmaxf((float)outdeg[rbase + r], 1.0f));
        h[(rbase + r) * D_OUT + n] = acc[r] * sc;
    }
}

// ---------------------------------------------------------------------------
// Kernel 4: SpMM scatter. One wave per edge: 32 lanes x float4 = 128 features.
// h (51 MB) and out (51 MB) are L2-resident on MI455X (192 MB L2), so both the
// gather and the f32 atomic adds stay on-chip.
// ---------------------------------------------------------------------------
__global__ __launch_bounds__(256) void gc_scatter_kernel(
        const float* __restrict__ h,
        const int* __restrict__ src,
        const int* __restrict__ dst,
        float* __restrict__ out) {
    const int lane = threadIdx.x & 31;
    const int e = blockIdx.x * 8 + (threadIdx.x >> 5);
    if (e >= N_EDGES) return;
    const int s = src[e];
    const int d = dst[e];
    const float4 v = *(const float4*)(h + (size_t)s * D_OUT + lane * 4);
    float* o = out + (size_t)d * D_OUT + lane * 4;
    atomicAdd(o + 0, v.x);
    atomicAdd(o + 1, v.y);
    atomicAdd(o + 2, v.z);
    atomicAdd(o + 3, v.w);
}

// ---------------------------------------------------------------------------
// Kernel 5: out = out * indeg^-1/2 + bias (in place).
// ---------------------------------------------------------------------------
__global__ void gc_finalize_kernel(float* __restrict__ out,
                                   const unsigned* __restrict__ indeg,
                                   const float* __restrict__ bias) {
    int i = blockIdx.x * blockDim.x + threadIdx.x;
    if (i >= N_NODES * D_OUT) return;
    int row = i >> 7;
    int f   = i & 127;
    float sc = 1.0f / sqrtf(fmaxf((float)indeg[row], 1.0f));
    out[i] = out[i] * sc + bias[f];
}

// ---------------------------------------------------------------------------
extern "C" void kernel_launch(void* const* d_in, const int* in_sizes, int n_in,
                              void* d_out, int out_size, void* d_ws, size_t ws_size,
                              hipStream_t stream) {
    const float* feat = (const float*)d_in[0];
    const float* W    = (const float*)d_in[1];
    const float* bias = (const float*)d_in[2];
    const int*   src  = (const int*)d_in[3];
    const int*   dst  = (const int*)d_in[4];
    float* out = (float*)d_out;

    // ws layout: [outdeg u32 x N][indeg u32 x N][h f32 x N*128]  (~52 MB)
    unsigned* outdeg = (unsigned*)d_ws;
    unsigned* indeg  = outdeg + N_NODES;
    float*    h      = (float*)(indeg + N_NODES);

    gc_zero_kernel<<<(N_NODES * D_OUT + 255) / 256, 256, 0, stream>>>(outdeg, indeg, out);
    gc_degree_kernel<<<(N_EDGES + 255) / 256, 256, 0, stream>>>(src, dst, outdeg, indeg);
    gc_gemm_kernel<<<N_NODES / 16, 256, 0, stream>>>(feat, W, outdeg, h);
    gc_scatter_kernel<<<(N_EDGES + 7) / 8, 256, 0, stream>>>(h, src, dst, out);
    gc_finalize_kernel<<<(N_NODES * D_OUT + 255) / 256, 256, 0, stream>>>(out, indeg, bias);
}